// GCN_18631568130050
// MI455X (gfx1250) — compile-verified
//
#include <hip/hip_runtime.h>
#include <math.h>

typedef __attribute__((ext_vector_type(2))) float v2f;
typedef __attribute__((ext_vector_type(8))) float v8f;

#define N_NODES 100000
#define N_EDGES 1600000
#define F_IN    256
#define H_DIM   128
#define C_DIM   40
#define C_PAD   48   // 3 tiles of 16

// ---------------------------------------------------------------------------
// 1) degree init (self-loop weight 1.0)
__global__ void k_init_deg(float* __restrict__ deg) {
    int i = blockIdx.x * blockDim.x + threadIdx.x;
    if (i < N_NODES) deg[i] = 1.0f;
}

// 2) accumulate edge weights into deg[col]
__global__ void k_accum_deg(const int* __restrict__ col,
                            const float* __restrict__ ea,
                            float* __restrict__ deg) {
    int e = blockIdx.x * blockDim.x + threadIdx.x;
    if (e < N_EDGES) atomicAdd(&deg[col[e]], ea[e]);
}

// 3) deg -> dinv = rsqrt(deg) in place
__global__ void k_dinv(float* __restrict__ deg) {
    int i = blockIdx.x * blockDim.x + threadIdx.x;
    if (i < N_NODES) {
        float d = deg[i];
        deg[i] = (d > 0.0f) ? rsqrtf(d) : 0.0f;
    }
}

// 4) per-edge norm = dinv[row] * ea * dinv[col]
__global__ void k_edge_norm(const int* __restrict__ row, const int* __restrict__ col,
                            const float* __restrict__ ea, const float* __restrict__ dinv,
                            float* __restrict__ norm) {
    int e = blockIdx.x * blockDim.x + threadIdx.x;
    if (e < N_EDGES) norm[e] = dinv[row[e]] * ea[e] * dinv[col[e]];
}

// ---------------------------------------------------------------------------
// GEMM1: h1[N,128] = x[N,256] @ W1[256,128] via V_WMMA_F32_16X16X4_F32.
// One wave per 16x16 output tile; 64 chained K=4 WMMAs.
__global__ void k_gemm1_wmma(const float* __restrict__ x,
                             const float* __restrict__ W1,
                             float* __restrict__ h1) {
    const int wave = threadIdx.x >> 5;
    const int lane = threadIdx.x & 31;
    const int tile = blockIdx.x * 8 + wave;
    const int TM = N_NODES / 16;   // 6250
    const int TN = H_DIM / 16;     // 8
    if (tile >= TM * TN) return;   // wave-uniform: EXEC stays all-ones
    const int tm = tile / TN;
    const int tn = tile % TN;
    const int half = lane >> 4;    // 0: K lo pair, 1: K hi pair
    const int l    = lane & 15;
    const int rowA = tm * 16 + l;  // this lane's M for the A fragment
    const int colB = tn * 16 + l;  // this lane's N for B / C / D fragments

    v8f c = {};
    for (int k0 = 0; k0 < F_IN; k0 += 4) {
        const int k = k0 + 2 * half;
        v2f a, b;
        a.x = x[(size_t)rowA * F_IN + k];
        a.y = x[(size_t)rowA * F_IN + k + 1];
        b.x = W1[(size_t)k       * H_DIM + colB];
        b.y = W1[(size_t)(k + 1) * H_DIM + colB];
        c = __builtin_amdgcn_wmma_f32_16x16x4_f32(false, a, false, b,
                                                  (short)0, c, false, false);
    }
    const int mBase = tm * 16 + half * 8;  // C/D layout: lanes 16-31 hold M=8..15
#pragma unroll
    for (int j = 0; j < 8; ++j)
        h1[(size_t)(mBase + j) * H_DIM + colB] = c[j];
}

// GEMM2: h2[N,40] = a1[N,128] @ W2[128,40] (N padded to 48 cols, guarded)
__global__ void k_gemm2_wmma(const float* __restrict__ a1,
                             const float* __restrict__ W2,
                             float* __restrict__ h2) {
    const int wave = threadIdx.x >> 5;
    const int lane = threadIdx.x & 31;
    const int tile = blockIdx.x * 8 + wave;
    const int TM = N_NODES / 16;    // 6250
    const int TN = C_PAD / 16;      // 3
    if (tile >= TM * TN) return;    // wave-uniform
    const int tm = tile / TN;
    const int tn = tile % TN;
    const int half = lane >> 4;
    const int l    = lane & 15;
    const int rowA = tm * 16 + l;
    const int n    = tn * 16 + l;
    const int ncl  = (n < C_DIM) ? n : (C_DIM - 1);   // clamp to stay in-bounds
    const float bm = (n < C_DIM) ? 1.0f : 0.0f;       // zero padded columns

    v8f c = {};
    for (int k0 = 0; k0 < H_DIM; k0 += 4) {
        const int k = k0 + 2 * half;
        v2f a, b;
        a.x = a1[(size_t)rowA * H_DIM + k];
        a.y = a1[(size_t)rowA * H_DIM + k + 1];
        b.x = W2[(size_t)k       * C_DIM + ncl] * bm;
        b.y = W2[(size_t)(k + 1) * C_DIM + ncl] * bm;
        c = __builtin_amdgcn_wmma_f32_16x16x4_f32(false, a, false, b,
                                                  (short)0, c, false, false);
    }
    const int mBase = tm * 16 + half * 8;
    if (n < C_DIM) {
#pragma unroll
        for (int j = 0; j < 8; ++j)
            h2[(size_t)(mBase + j) * C_DIM + n] = c[j];
    }
}

// ---------------------------------------------------------------------------
// 6) out1[i,f] = b1[f] + dinv[i]^2 * h1[i,f]   (bias + self-loop term)
__global__ void k_init_out1(const float* __restrict__ h1, const float* __restrict__ b1,
                            const float* __restrict__ dinv, float* __restrict__ out1) {
    size_t tid = (size_t)blockIdx.x * blockDim.x + threadIdx.x;
    if (tid >= (size_t)N_NODES * H_DIM) return;
    int i = (int)(tid >> 7);
    int f = (int)(tid & 127);
    float di = dinv[i];
    out1[tid] = b1[f] + di * di * h1[tid];
}

// 7) edge scatter, layer 1: 32 lanes per edge, float4 gather, 4 atomics/lane
__global__ void k_scatter1(const float* __restrict__ h1,
                           const int* __restrict__ row, const int* __restrict__ col,
                           const float* __restrict__ norm, float* __restrict__ out1) {
    size_t tid = (size_t)blockIdx.x * blockDim.x + threadIdx.x;
    int e  = (int)(tid >> 5);
    int f4 = (int)(tid & 31) * 4;
    if (e >= N_EDGES) return;
    int r = row[e], cdst = col[e];
    float w = norm[e];
    const float4 hv = *(const float4*)(h1 + (size_t)r * H_DIM + f4);
    float* dst = out1 + (size_t)cdst * H_DIM + f4;
    atomicAdd(dst + 0, hv.x * w);
    atomicAdd(dst + 1, hv.y * w);
    atomicAdd(dst + 2, hv.z * w);
    atomicAdd(dst + 3, hv.w * w);
}

// 8) ReLU in place
__global__ void k_relu(float* __restrict__ a) {
    size_t tid = (size_t)blockIdx.x * blockDim.x + threadIdx.x;
    if (tid < (size_t)N_NODES * H_DIM) a[tid] = fmaxf(a[tid], 0.0f);
}

// 10) out2[i,c] = b2[c] + dinv[i]^2 * h2[i,c]   (writes directly into d_out)
__global__ void k_init_out2(const float* __restrict__ h2, const float* __restrict__ b2,
                            const float* __restrict__ dinv, float* __restrict__ out2) {
    size_t tid = (size_t)blockIdx.x * blockDim.x + threadIdx.x;
    if (tid >= (size_t)N_NODES * C_DIM) return;
    int i = (int)(tid / C_DIM);
    int f = (int)(tid % C_DIM);
    float di = dinv[i];
    out2[tid] = b2[f] + di * di * h2[tid];
}

// 11) edge scatter, layer 2: 64 lanes per edge slot, f<40 active
__global__ void k_scatter2(const float* __restrict__ h2,
                           const int* __restrict__ row, const int* __restrict__ col,
                           const float* __restrict__ norm, float* __restrict__ out2) {
    size_t tid = (size_t)blockIdx.x * blockDim.x + threadIdx.x;
    int e = (int)(tid >> 6);
    int f = (int)(tid & 63);
    if (e >= N_EDGES || f >= C_DIM) return;
    float w = norm[e];
    float v = h2[(size_t)row[e] * C_DIM + f] * w;
    atomicAdd(&out2[(size_t)col[e] * C_DIM + f], v);
}

// 12) log_softmax over 40 classes, in place on d_out (one thread per node row)
__global__ void k_log_softmax(float* __restrict__ out) {
    int i = blockIdx.x * blockDim.x + threadIdx.x;
    if (i >= N_NODES) return;
    float* rowp = out + (size_t)i * C_DIM;
    float v[C_DIM];
    float mx = -INFINITY;
#pragma unroll
    for (int c = 0; c < C_DIM; ++c) { v[c] = rowp[c]; mx = fmaxf(mx, v[c]); }
    float s = 0.0f;
#pragma unroll
    for (int c = 0; c < C_DIM; ++c) s += expf(v[c] - mx);
    float lse = mx + logf(s);
#pragma unroll
    for (int c = 0; c < C_DIM; ++c) rowp[c] = v[c] - lse;
}

// ---------------------------------------------------------------------------
extern "C" void kernel_launch(void* const* d_in, const int* in_sizes, int n_in,
                              void* d_out, int out_size, void* d_ws, size_t ws_size,
                              hipStream_t stream) {
    const float* x   = (const float*)d_in[0];
    const int*   ei  = (const int*)  d_in[1];     // [2, E]
    const float* ea  = (const float*)d_in[2];
    const float* W1  = (const float*)d_in[3];
    const float* b1  = (const float*)d_in[4];
    const float* W2  = (const float*)d_in[5];
    const float* b2  = (const float*)d_in[6];
    const int* row = ei;
    const int* col = ei + N_EDGES;
    float* out = (float*)d_out;

    // workspace layout (floats)
    float* ws   = (float*)d_ws;
    float* dinv = ws;                                   // N       (deg then dinv)
    float* norm = dinv + N_NODES;                       // E
    float* h1   = norm + N_EDGES;                       // N*128
    float* out1 = h1   + (size_t)N_NODES * H_DIM;       // N*128
    float* h2   = out1 + (size_t)N_NODES * H_DIM;       // N*40

    const int B = 256;
    // normalization
    k_init_deg  <<<(N_NODES + B - 1) / B, B, 0, stream>>>(dinv);
    k_accum_deg <<<(N_EDGES + B - 1) / B, B, 0, stream>>>(col, ea, dinv);
    k_dinv      <<<(N_NODES + B - 1) / B, B, 0, stream>>>(dinv);
    k_edge_norm <<<(N_EDGES + B - 1) / B, B, 0, stream>>>(row, col, ea, dinv, norm);

    // layer 1
    k_gemm1_wmma<<<(N_NODES / 16) * (H_DIM / 16) / 8, B, 0, stream>>>(x, W1, h1);
    k_init_out1 <<<(int)(((size_t)N_NODES * H_DIM + B - 1) / B), B, 0, stream>>>(h1, b1, dinv, out1);
    k_scatter1  <<<(int)(((size_t)N_EDGES * 32 + B - 1) / B), B, 0, stream>>>(h1, row, col, norm, out1);
    k_relu      <<<(int)(((size_t)N_NODES * H_DIM + B - 1) / B), B, 0, stream>>>(out1);

    // layer 2
    {
        int tiles = (N_NODES / 16) * (C_PAD / 16);
        k_gemm2_wmma<<<(tiles + 7) / 8, B, 0, stream>>>(out1, W2, h2);
    }
    k_init_out2 <<<(int)(((size_t)N_NODES * C_DIM + B - 1) / B), B, 0, stream>>>(h2, b2, dinv, out);
    k_scatter2  <<<(int)(((size_t)N_EDGES * 64 + B - 1) / B), B, 0, stream>>>(h2, row, col, norm, out);

    // log-softmax
    k_log_softmax<<<(N_NODES + B - 1) / B, B, 0, stream>>>(out);
}